// PointAttentionEncoder1D_41532333752462
// MI455X (gfx1250) — compile-verified
//
#include <hip/hip_runtime.h>
#include <math.h>

// ---------------------------------------------------------------------------
// PointAttentionEncoder1D fused kernel for gfx1250 (CDNA5, wave32).
// All linear layers run on V_WMMA_F32_16X16X4_F32 (fp32 matches reference,
// K=4 tiles fit the tiny channel dims 3/16/32/64 exactly).
// ---------------------------------------------------------------------------

typedef __attribute__((ext_vector_type(2))) float v2f;
typedef __attribute__((ext_vector_type(8))) float v8f;

#define NPB 8              // points per workgroup
#define RV  (NPB * 17)     // valid rows = 136
#define RP  144            // rows padded to multiple of 16
#define NTI (RP / 16)      // 9 row tiles
#define ST  33             // LDS row stride for [R][32] matrices (bank-spread)
#define HST 65             // LDS row stride for 16x64 hidden scratch

__device__ __forceinline__ v8f wmma4(v2f a, v2f b, v8f c) {
  // D = A(16x4) * B(4x16) + C(16x16), fp32
  return __builtin_amdgcn_wmma_f32_16x16x4_f32(
      /*neg_a=*/false, a, /*neg_b=*/false, b,
      /*c_mod=*/(short)0, c, /*reuse_a=*/false, /*reuse_b=*/false);
}

// A fragment: 16x4 tile, rows r0..r0+15, cols k0..k0+3 from LDS (row-major, ld floats)
// lane 0-15 -> M=lane, K={k0,k0+1}; lane 16-31 -> M=lane-16, K={k0+2,k0+3}
__device__ __forceinline__ v2f afrag(const float* base, int ld, int r0, int k0, int lane) {
  int m = lane & 15, half = lane >> 4;
  const float* p = base + (r0 + m) * ld + k0 + 2 * half;
  v2f a; a.x = p[0]; a.y = p[1];
  return a;
}

// B fragment: 4x16 tile of W[K][N] (row-major, ldn floats), rows k0.., cols n0..
// lane 0-15 -> col n=lane, K={k0,k0+1}; lane 16-31 -> col n=lane-16, K={k0+2,k0+3}
__device__ __forceinline__ v2f bfrag(const float* w, int ldn, int k0, int n0, int lane) {
  int n = lane & 15, half = lane >> 4;
  v2f b;
  b.x = w[(k0 + 2 * half + 0) * ldn + n0 + n];
  b.y = w[(k0 + 2 * half + 1) * ldn + n0 + n];
  return b;
}

// C/D store: VGPR i holds row (r0 + 8*half + i), col n0 + (lane&15)
__device__ __forceinline__ void cstore(float* base, int ld, int r0, int n0, int lane, v8f c) {
  int n = lane & 15, half = lane >> 4;
#pragma unroll
  for (int i = 0; i < 8; ++i)
    base[(r0 + 8 * half + i) * ld + n0 + n] = c[i];
}

__device__ __forceinline__ v8f splat8(float v) {
  v8f r;
#pragma unroll
  for (int i = 0; i < 8; ++i) r[i] = v;
  return r;
}

__device__ __forceinline__ v8f lrelu8(v8f h) {
#pragma unroll
  for (int i = 0; i < 8; ++i) {
    float x = h[i];
    h[i] = (x > 0.0f) ? x : 0.01f * x;  // jax.nn.leaky_relu default slope
  }
  return h;
}

__global__ __launch_bounds__(256) void pae_fused_kernel(
    const float* __restrict__ center, const float* __restrict__ other,
    const float* __restrict__ mlp_w1, const float* __restrict__ mlp_b1,
    const float* __restrict__ mlp_w2, const float* __restrict__ mlp_b2,
    const float* __restrict__ wp_w1, const float* __restrict__ wp_b1,
    const float* __restrict__ wp_w2, const float* __restrict__ wp_b2,
    float* __restrict__ out, int P) {
  // ---- LDS (~131 KB) ----
  __shared__ float sW1P[6 * 4 * 16];   // [6][4][16]  w1 with bias folded as row 3
  __shared__ float sW2[6 * 16 * 32];   // [6][16][32]
  __shared__ float sB2[6 * 32];        // [6][32]
  __shared__ float sWP1[32 * 64];
  __shared__ float sWPB1[64];
  __shared__ float sWP2[64 * 32];
  __shared__ float sWPB2[32];
  __shared__ float sCEN[16 * 4];       // (cx,cy,cz,1) per point, padded to 16 rows
  __shared__ float sALL[RP * 4];       // (x,y,z,1) per row
  __shared__ float sDM[RP * 4];        // (cx-x,cy-y,cz-z,1) per row
  __shared__ float sQO[16 * ST];       // q per point
  __shared__ float sCKO[16 * ST];      // ck per point
  __shared__ float sCVO[16 * ST];      // cv per point
  __shared__ float sK[RP * ST];        // ok ...then X = q - k + pos (in place)
  __shared__ float sV[RP * ST];        // ov ...then VP = v + pos (in place)
  __shared__ float sPOS[RP * ST];      // pos ...then FW = weight_process(X) (in place)
  __shared__ float sHS[8 * 16 * HST];  // per-wave 16x64 hidden scratch
  __shared__ float sRMAX[RP];
  __shared__ float sRSUM[RP];

  const int tid = threadIdx.x;
  const int wave = tid >> 5;
  const int lane = tid & 31;
  const int p0 = blockIdx.x * NPB;

  // ---- Phase 1: stage weights ----
  for (int i = tid; i < 6 * 4 * 16; i += 256) {
    int m = i >> 6, k = (i >> 4) & 3, n = i & 15;
    sW1P[i] = (k < 3) ? mlp_w1[m * 48 + k * 16 + n] : mlp_b1[m * 16 + n];
  }
  for (int i = tid; i < 6 * 16 * 32; i += 256) sW2[i] = mlp_w2[i];
  if (tid < 6 * 32) sB2[tid] = mlp_b2[tid];
  for (int i = tid; i < 32 * 64; i += 256) sWP1[i] = wp_w1[i];
  if (tid < 64) sWPB1[tid] = wp_b1[tid];
  for (int i = tid; i < 64 * 32; i += 256) sWP2[i] = wp_w2[i];
  if (tid < 32) sWPB2[tid] = wp_b2[tid];

  // ---- Phase 2: build input matrices ----
  for (int r = tid; r < RP; r += 256) {
    int pl = r / 17, j = r % 17;
    float cx = 0.f, cy = 0.f, cz = 0.f, x = 0.f, y = 0.f, z = 0.f;
    if (r < RV && (p0 + pl) < P) {
      const float* cp = center + (size_t)(p0 + pl) * 3;
      cx = cp[0]; cy = cp[1]; cz = cp[2];
      if (j == 0) { x = cx; y = cy; z = cz; }
      else {
        const float* op = other + ((size_t)(p0 + pl) * 16 + (j - 1)) * 3;
        x = op[0]; y = op[1]; z = op[2];
      }
    }
    sALL[r * 4 + 0] = x;      sALL[r * 4 + 1] = y;
    sALL[r * 4 + 2] = z;      sALL[r * 4 + 3] = 1.0f;
    sDM[r * 4 + 0] = cx - x;  sDM[r * 4 + 1] = cy - y;
    sDM[r * 4 + 2] = cz - z;  sDM[r * 4 + 3] = 1.0f;
  }
  if (tid < 16) {
    float cx = 0.f, cy = 0.f, cz = 0.f;
    if (tid < NPB && (p0 + tid) < P) {
      const float* cp = center + (size_t)(p0 + tid) * 3;
      cx = cp[0]; cy = cp[1]; cz = cp[2];
    }
    sCEN[tid * 4 + 0] = cx; sCEN[tid * 4 + 1] = cy;
    sCEN[tid * 4 + 2] = cz; sCEN[tid * 4 + 3] = 1.0f;
  }
  __syncthreads();

  // ---- Phase 3: small MLPs (3->16->32) as WMMA tile-jobs ----
  // jobs 0..2: q/ck/cv on the 16-row center matrix; jobs 3..29: ok/ov/pos tiles.
  const int NJOBS = 3 + 3 * NTI;
  float* hs = &sHS[wave * 16 * HST];
  for (int jb = wave; jb < NJOBS; jb += 8) {
    int m, r0;
    const float* src;
    float* dst;
    if (jb < 3) {
      m = jb; r0 = 0; src = sCEN;
      dst = (jb == 0) ? sQO : (jb == 1) ? sCKO : sCVO;
    } else {
      int t = jb - 3, mm = t % 3;
      m = 3 + mm; r0 = (t / 3) * 16;
      src = (mm == 2) ? sDM : sALL;
      dst = (mm == 0) ? sK : (mm == 1) ? sV : sPOS;
    }
    // layer 1: [16x4] x [4x16] with bias folded; one WMMA
    v2f a1 = afrag(src, 4, (jb < 3) ? 0 : r0, 0, lane);
    v2f b1 = bfrag(&sW1P[m * 64], 16, 0, 0, lane);
    v8f h = wmma4(a1, b1, splat8(0.0f));
    h = lrelu8(h);
    cstore(hs, HST, 0, 0, lane, h);  // re-fragment via per-wave LDS scratch
    // layer 2: [16x16] x [16x32]; 2 n-tiles x 4 k-steps
    const float* w2 = &sW2[m * 512];
    const float* b2 = &sB2[m * 32];
#pragma unroll
    for (int nt = 0; nt < 2; ++nt) {
      v8f c = splat8(b2[nt * 16 + (lane & 15)]);
#pragma unroll
      for (int ks = 0; ks < 4; ++ks) {
        v2f a = afrag(hs, HST, 0, ks * 4, lane);
        v2f b = bfrag(w2, 32, ks * 4, nt * 16, lane);
        c = wmma4(a, b, c);
      }
      cstore(dst, ST, r0, nt * 16, lane, c);
    }
  }
  __syncthreads();

  // ---- Phase 4: assemble X = q - k + pos and VP = v + pos (in place) ----
  for (int e = tid; e < RV * 32; e += 256) {
    int r = e >> 5, c = e & 31;
    int pl = r / 17, j = r % 17;
    float pos = sPOS[r * ST + c];
    float kv = (j == 0) ? sCKO[pl * ST + c] : sK[r * ST + c];
    float vv = (j == 0) ? sCVO[pl * ST + c] : sV[r * ST + c];
    float q = sQO[pl * ST + c];
    sK[r * ST + c] = q - kv + pos;  // X
    sV[r * ST + c] = vv + pos;      // VP
  }
  __syncthreads();

  // ---- Phase 5: weight_process 32->64->32 per 16-row tile ----
  for (int t = wave; t < NTI; t += 8) {
    int r0 = t * 16;
    v8f h[4];
#pragma unroll
    for (int nt = 0; nt < 4; ++nt) h[nt] = splat8(sWPB1[nt * 16 + (lane & 15)]);
#pragma unroll
    for (int ks = 0; ks < 8; ++ks) {
      v2f a = afrag(sK, ST, r0, ks * 4, lane);
#pragma unroll
      for (int nt = 0; nt < 4; ++nt) {
        v2f b = bfrag(sWP1, 64, ks * 4, nt * 16, lane);
        h[nt] = wmma4(a, b, h[nt]);
      }
    }
#pragma unroll
    for (int nt = 0; nt < 4; ++nt) {
      h[nt] = lrelu8(h[nt]);
      cstore(hs, HST, 0, nt * 16, lane, h[nt]);
    }
#pragma unroll
    for (int nt = 0; nt < 2; ++nt) {
      v8f c = splat8(sWPB2[nt * 16 + (lane & 15)]);
#pragma unroll
      for (int ks = 0; ks < 16; ++ks) {
        v2f a = afrag(hs, HST, 0, ks * 4, lane);
        v2f b = bfrag(sWP2, 32, ks * 4, nt * 16, lane);
        c = wmma4(a, b, c);
      }
      cstore(sPOS, ST, r0, nt * 16, lane, c);  // FW overwrites pos buffer
    }
  }
  __syncthreads();

  // ---- Phase 6: per-row softmax stats over 32 channels ----
  for (int r = tid; r < RV; r += 256) {
    float mx = -3.402823466e38f;
    for (int c = 0; c < 32; ++c) mx = fmaxf(mx, sPOS[r * ST + c]);
    float s = 0.0f;
    for (int c = 0; c < 32; ++c) s += __expf(sPOS[r * ST + c] - mx);
    sRMAX[r] = mx;
    sRSUM[r] = 1.0f / s;
  }
  __syncthreads();

  // ---- Phase 7: attn = softmax(FW) * VP, summed over the 17 rows ----
  for (int e = tid; e < NPB * 32; e += 256) {  // exactly one element per thread
    int pl = e >> 5, c = e & 31;
    int pg = p0 + pl;
    if (pg < P) {
      float acc = 0.0f;
#pragma unroll
      for (int j = 0; j < 17; ++j) {
        int r = pl * 17 + j;
        acc += __expf(sPOS[r * ST + c] - sRMAX[r]) * sRSUM[r] * sV[r * ST + c];
      }
      out[(size_t)pg * 32 + c] = acc;
    }
  }
}

extern "C" void kernel_launch(void* const* d_in, const int* in_sizes, int n_in,
                              void* d_out, int out_size, void* d_ws, size_t ws_size,
                              hipStream_t stream) {
  const float* center = (const float*)d_in[0];
  const float* other  = (const float*)d_in[1];
  const float* mlp_w1 = (const float*)d_in[2];
  const float* mlp_b1 = (const float*)d_in[3];
  const float* mlp_w2 = (const float*)d_in[4];
  const float* mlp_b2 = (const float*)d_in[5];
  const float* wp_w1  = (const float*)d_in[6];
  const float* wp_b1  = (const float*)d_in[7];
  const float* wp_w2  = (const float*)d_in[8];
  const float* wp_b2  = (const float*)d_in[9];
  float* out = (float*)d_out;

  int P = in_sizes[0] / 3;  // center is [P,1,3]
  int blocks = (P + NPB - 1) / NPB;
  hipLaunchKernelGGL(pae_fused_kernel, dim3(blocks), dim3(256), 0, stream,
                     center, other, mlp_w1, mlp_b1, mlp_w2, mlp_b2,
                     wp_w1, wp_b1, wp_w2, wp_b2, out, P);
}